// HierarchicalHeteroGraphSage_7954279432525
// MI455X (gfx1250) — compile-verified
//
#include <hip/hip_runtime.h>

#define N_NODES 100000
#define E_EDGES 800000
#define D 128
#define NLAYERS 3
#define NREL 3
#define KTOT (4 * D)    // 512: {mean0, mean1, mean2, h} concat along K
#define KCHUNK 32       // k-rows per LDS chunk (16 KB), double-buffered
#define NCHUNK (KTOT / KCHUNK)
#define NSTEP (KCHUNK / 4)   // 8 wmma k-steps per chunk

typedef __attribute__((ext_vector_type(2))) float v2f;
typedef __attribute__((ext_vector_type(8))) float v8f;
typedef int b128_t __attribute__((vector_size(16)));   // matches builtin param type

#define GLOBAL_AS __attribute__((address_space(1)))
#define LDS_AS    __attribute__((address_space(3)))

#if defined(__HIP_DEVICE_COMPILE__) && \
    __has_builtin(__builtin_amdgcn_global_load_async_to_lds_b128) && \
    __has_builtin(__builtin_amdgcn_s_wait_asynccnt)
#define USE_ASYNC_LDS 1
#else
#define USE_ASYNC_LDS 0
#endif

// ---------------------------------------------------------------- zero
__global__ void zero_kernel(float* __restrict__ p, long long n) {
  long long i = (long long)blockIdx.x * blockDim.x + threadIdx.x;
  if (i < n) p[i] = 0.0f;
}

// ---------------------------------------------------------------- edge aggregation
// One wave (32 lanes) per edge; lane covers 4 consecutive floats of the D=128 row.
// Coalesced 512B gather of h[src], native f32 atomics into L2-resident agg[dst].
__global__ void agg_edges_kernel(const float* __restrict__ h,
                                 const int* __restrict__ ei,     // [2, E]
                                 const int* __restrict__ emask,  // [E]
                                 int layer,
                                 float* __restrict__ agg,        // [N, D]
                                 float* __restrict__ cnt) {      // [N]
  int wid  = blockIdx.x * (blockDim.x >> 5) + (threadIdx.x >> 5);
  int lane = threadIdx.x & 31;
  if (wid >= E_EDGES) return;
  if (emask[wid] > layer) return;          // edge trimmed this layer
  int src = ei[wid];
  int dst = ei[E_EDGES + wid];
  const float4* hrow = (const float4*)(h + (size_t)src * D);
  float4 v = hrow[lane];
  float* arow = agg + (size_t)dst * D + lane * 4;
  unsafeAtomicAdd(arow + 0, v.x);
  unsafeAtomicAdd(arow + 1, v.y);
  unsafeAtomicAdd(arow + 2, v.z);
  unsafeAtomicAdd(arow + 3, v.w);
  if (lane == 0) unsafeAtomicAdd(cnt + dst, 1.0f);
}

// ---------------------------------------------------------------- agg -> mean (in place)
__global__ void mean_kernel(float* __restrict__ agg, const float* __restrict__ cnt) {
  int wid  = blockIdx.x * (blockDim.x >> 5) + (threadIdx.x >> 5);  // over 3*N rows
  int lane = threadIdx.x & 31;
  if (wid >= NREL * N_NODES) return;
  float inv = 1.0f / fmaxf(cnt[wid], 1.0f);
  float4* row = (float4*)(agg + (size_t)wid * D);
  float4 v = row[lane];
  v.x *= inv; v.y *= inv; v.z *= inv; v.w *= inv;
  row[lane] = v;
}

// ---------------------------------------------------------------- weight prepack
// WtI holds the GEMM B-matrix ALREADY in the WMMA-friendly k-pair-interleaved
// layout:  WtI[(k>>1)*256 + o*2 + (k&1)] = Wt[k][o]
// where Wt[k][o]: rows 0..383 = W_l[i, k/128]^T, rows 384..511 = (sum_t W_r[i,t])^T.
// This makes every 32-k chunk a CONTIGUOUS 16 KB span -> straight async copy to LDS.
__global__ void prep_kernel(const float* __restrict__ W_l,
                            const float* __restrict__ W_r,
                            const float* __restrict__ b_l,
                            int li, float* __restrict__ WtI, float* __restrict__ bias) {
  int idx = blockIdx.x * blockDim.x + threadIdx.x;  // 512*128
  if (idx >= KTOT * D) return;
  int k = idx / D, o = idx % D;
  float v;
  if (k < 3 * D) {
    int t = k >> 7, kk = k & (D - 1);
    v = W_l[(((size_t)li * NREL + t) * D + o) * D + kk];
  } else {
    int kk = k - 3 * D;
    v = 0.0f;
    for (int t = 0; t < NREL; ++t)
      v += W_r[(((size_t)li * NREL + t) * D + o) * D + kk];
  }
  WtI[((size_t)(k >> 1) * (2 * D)) + o * 2 + (k & 1)] = v;
  if (idx < D) {
    float b = 0.0f;
    for (int t = 0; t < NREL; ++t) b += b_l[((size_t)li * NREL + t) * D + idx];
    bias[idx] = b;
  }
}

// ---------------------------------------------------------------- async chunk copy
// Copies one contiguous 16 KB chunk of WtI into an LDS buffer.
// 256 threads x 4 x b128 = 16 KB; 4 async instructions per wave.
__device__ __forceinline__ void issue_chunk_copy(const float* __restrict__ src,
                                                 float* __restrict__ dst, int tid) {
#if USE_ASYNC_LDS
  GLOBAL_AS b128_t* g = (GLOBAL_AS b128_t*)(void*)const_cast<float*>(src);
  LDS_AS b128_t*    l = (LDS_AS b128_t*)(void*)dst;
#pragma unroll
  for (int j = 0; j < 4; ++j) {
    __builtin_amdgcn_global_load_async_to_lds_b128(
        g + tid + j * 256, l + tid + j * 256, 0, 0);
  }
#else
  const float4* s4 = (const float4*)src;
  float4* d4 = (float4*)dst;
#pragma unroll
  for (int j = 0; j < 4; ++j) d4[tid + j * 256] = s4[tid + j * 256];
#endif
}

// ---------------------------------------------------------------- fused GEMM + bias + mask + relu
// C[N,128] = X[N,512] @ Wt[512,128];  X = {mean0 | mean1 | mean2 | hin} (virtual concat).
// Block = 256 threads (8 waves) -> 128x128 output tile; wave = 16 rows x 128 cols
// (8 accumulators of v8f).  Weights streamed chunk-by-chunk into double-buffered
// LDS via async global->LDS copies (ASYNCcnt), overlapped with WMMA on the
// previous chunk.  B fragments are software-pipelined in registers: step s+1's
// 8 ds_load_b64 issue while step s's 8 WMMAs execute.
__global__ __launch_bounds__(256)
void gemm_kernel(const float* __restrict__ hin,    // [N, D]
                 const float* __restrict__ mean3,  // [3, N, D]
                 const float* __restrict__ WtI,    // [512, D] pair-interleaved
                 const float* __restrict__ bias,   // [D]
                 const int* __restrict__ node_mask,
                 int layer,
                 float* __restrict__ hout) {       // [N, D] (may alias hin)
  __shared__ float sB[2][KCHUNK * D];  // 2 x 16 KB double buffer

  const int tid  = threadIdx.x;
  const int wave = tid >> 5;
  const int lane = tid & 31;
  const int half = lane >> 4;   // 0: K group / M group 0, 1: K+2 / M+8 group
  const int l16  = lane & 15;
  const int r0   = blockIdx.x * 128 + wave * 16;
  const int arow = r0 + l16;                  // A-matrix row this lane feeds
  const bool avalid = (arow < N_NODES);

  v8f acc[8] = {};

#if USE_ASYNC_LDS
  issue_chunk_copy(WtI + 0 * KCHUNK * D, sB[0], tid);
  issue_chunk_copy(WtI + 1 * KCHUNK * D, sB[1], tid);
#endif

  for (int c = 0; c < NCHUNK; ++c) {
    float* sBc = sB[c & 1];

#if USE_ASYNC_LDS
    // chunk c complete (keep chunk c+1's 4 ops in flight), then publish to block
    if (c + 1 < NCHUNK) __builtin_amdgcn_s_wait_asynccnt(4);
    else                __builtin_amdgcn_s_wait_asynccnt(0);
    __syncthreads();
#else
    __syncthreads();                       // buffer reuse fence
    issue_chunk_copy(WtI + (size_t)c * KCHUNK * D, sBc, tid);
    __syncthreads();
#endif

    // ---- source matrix for this K-chunk of the virtual concat ----
    const int srcSel = c >> 2;             // 0,1,2 -> mean rel; 3 -> hin
    const int kkBase = (c & 3) * KCHUNK;
    const float* Xsrc = (srcSel < 3) ? (mean3 + (size_t)srcSel * N_NODES * D) : hin;
    const float* Xrow = Xsrc + (size_t)arow * D + kkBase;

    // ---- preload all A fragments of this chunk (one load clause) ----
    v2f afr[NSTEP];
#pragma unroll
    for (int s = 0; s < NSTEP; ++s) {
      v2f a = {0.0f, 0.0f};
      if (avalid) a = *(const v2f*)(Xrow + s * 4 + half * 2);
      afr[s] = a;
    }

    // ---- WMMA with register-pipelined B fragments ----
    v2f bfr[8], bnx[8];
#pragma unroll
    for (int ct = 0; ct < 8; ++ct)
      bfr[ct] = *(const v2f*)(&sBc[(0 * 2 + half) * (2 * D) + (ct * 16 + l16) * 2]);
#pragma unroll
    for (int s = 0; s < NSTEP; ++s) {
      if (s + 1 < NSTEP) {
#pragma unroll
        for (int ct = 0; ct < 8; ++ct)
          bnx[ct] = *(const v2f*)(&sBc[((s + 1) * 2 + half) * (2 * D) + (ct * 16 + l16) * 2]);
      }
#pragma unroll
      for (int ct = 0; ct < 8; ++ct) {
        acc[ct] = __builtin_amdgcn_wmma_f32_16x16x4_f32(
            false, afr[s], false, bfr[ct], (short)0, acc[ct], false, false);
      }
#pragma unroll
      for (int ct = 0; ct < 8; ++ct) bfr[ct] = bnx[ct];
    }

#if USE_ASYNC_LDS
    __syncthreads();                       // everyone done reading sB[c&1]
    if (c + 2 < NCHUNK)
      issue_chunk_copy(WtI + (size_t)(c + 2) * KCHUNK * D, sBc, tid);
#endif
  }

  // ---- epilogue: + bias, node-mask, relu, store ----
  bool rowok[8];
  bool nvalid[8];
#pragma unroll
  for (int v = 0; v < 8; ++v) {
    int row = r0 + v + half * 8;     // C layout: VGPR v -> M = v (+8 for upper half)
    rowok[v]  = (row < N_NODES);
    nvalid[v] = rowok[v] && (node_mask[row] <= layer);
  }
#pragma unroll
  for (int ct = 0; ct < 8; ++ct) {
    int col = ct * 16 + l16;
    float bsum = bias[col];
#pragma unroll
    for (int v = 0; v < 8; ++v) {
      if (rowok[v]) {
        int row = r0 + v + half * 8;
        float val = nvalid[v] ? fmaxf(acc[ct][v] + bsum, 0.0f) : 0.0f;
        hout[(size_t)row * D + col] = val;
      }
    }
  }
}

// ---------------------------------------------------------------- host driver
extern "C" void kernel_launch(void* const* d_in, const int* in_sizes, int n_in,
                              void* d_out, int out_size, void* d_ws, size_t ws_size,
                              hipStream_t stream) {
  (void)in_sizes; (void)n_in; (void)out_size; (void)ws_size;
  const float* x      = (const float*)d_in[0];
  const int* ei[3]    = {(const int*)d_in[1], (const int*)d_in[2], (const int*)d_in[3]};
  const int* nodem    = (const int*)d_in[4];
  const int* em[3]    = {(const int*)d_in[5], (const int*)d_in[6], (const int*)d_in[7]};
  const float* W_l    = (const float*)d_in[8];
  const float* b_l    = (const float*)d_in[9];
  const float* W_r    = (const float*)d_in[10];

  // workspace layout
  float* agg  = (float*)d_ws;                           // 3*N*D
  float* cnt  = agg  + (size_t)NREL * N_NODES * D;      // 3*N
  float* hbuf = cnt  + (size_t)NREL * N_NODES;          // N*D
  float* WtI  = hbuf + (size_t)N_NODES * D;             // 512*D (interleaved)
  float* bias = WtI  + (size_t)KTOT * D;                // D

  const long long zlen = (long long)NREL * N_NODES * D + (long long)NREL * N_NODES;

  for (int i = 0; i < NLAYERS; ++i) {
    const int layer = NLAYERS - i;                      // trim_to_layer semantics
    const float* hin = (i == 0) ? x : hbuf;
    float* hout = (i == NLAYERS - 1) ? (float*)d_out : hbuf;  // in-place update is safe

    zero_kernel<<<(int)((zlen + 1023) / 1024), 1024, 0, stream>>>(agg, zlen);
    for (int t = 0; t < NREL; ++t) {
      agg_edges_kernel<<<E_EDGES / 8, 256, 0, stream>>>(
          hin, ei[t], em[t], layer,
          agg + (size_t)t * N_NODES * D, cnt + (size_t)t * N_NODES);
    }
    mean_kernel<<<(NREL * N_NODES + 7) / 8, 256, 0, stream>>>(agg, cnt);
    prep_kernel<<<(KTOT * D + 255) / 256, 256, 0, stream>>>(W_l, W_r, b_l, i, WtI, bias);
    gemm_kernel<<<(N_NODES + 127) / 128, 256, 0, stream>>>(
        hin, agg, WtI, bias, nodem, layer, hout);
  }
}